// LATTE_55078660604154
// MI455X (gfx1250) — compile-verified
//
#include <hip/hip_runtime.h>

#define NN 100000      // nodes
#define FF 256         // in features
#define DD 128         // hidden
#define HH 4           // heads
#define CC 32          // channels per head
#define EE 1000000     // edges per metapath

typedef float v2f __attribute__((ext_vector_type(2)));
typedef float v8f __attribute__((ext_vector_type(8)));

// ---------------- helpers ----------------

__device__ __forceinline__ void atomicAddF32(float* p, float v) {
    __hip_atomic_fetch_add(p, v, __ATOMIC_RELAXED, __HIP_MEMORY_SCOPE_AGENT);
}

// monotonic float <-> uint mapping so atomicMax(u32) == float max
__device__ __forceinline__ unsigned enc_f32(float f) {
    unsigned u = __float_as_uint(f);
    return (u & 0x80000000u) ? ~u : (u | 0x80000000u);
}
__device__ __forceinline__ float dec_f32(unsigned u) {
    unsigned b = (u & 0x80000000u) ? (u & 0x7FFFFFFFu) : ~u;
    return __uint_as_float(b);
}

// ---------------- kernels ----------------

__global__ void zero_u32(unsigned* __restrict__ p, long n) {
    long i = (long)blockIdx.x * blockDim.x + threadIdx.x;
    if (i < n) p[i] = 0u;
}

// h = x @ W_lin + b_lin   via V_WMMA_F32_16X16X4_F32
// grid.x = NN/16 row tiles; 256 threads = 8 waves, wave w owns cols [16w,16w+16)
__global__ __launch_bounds__(256) void gemm_h(const float* __restrict__ x,
                                              const float* __restrict__ W,
                                              const float* __restrict__ bias,
                                              float* __restrict__ h) {
    __shared__ float xs[16][FF + 4];   // +4 pad: column reads hit distinct banks
    const int row0 = blockIdx.x * 16;
    #pragma unroll
    for (int i = 0; i < 16; ++i)       // coalesced: 256 threads load one row each pass
        xs[i][threadIdx.x] = x[(long)(row0 + i) * FF + threadIdx.x];
    __syncthreads();

    const int wave = threadIdx.x >> 5;
    const int lane = threadIdx.x & 31;
    const int mn   = lane & 15;        // A: row m   B: col n   D: col n
    const int half = lane >> 4;
    const int col0 = wave * 16;

    v8f acc = {};
    for (int k0 = 0; k0 < FF; k0 += 4) {
        const int ka = k0 + 2 * half;
        v2f a, b;
        a.x = xs[mn][ka];
        a.y = xs[mn][ka + 1];
        b.x = W[(long)ka * DD + col0 + mn];
        b.y = W[(long)(ka + 1) * DD + col0 + mn];
        acc = __builtin_amdgcn_wmma_f32_16x16x4_f32(false, a, false, b,
                                                    (short)0, acc, false, false);
    }
    const float bv = bias[col0 + mn];
    #pragma unroll
    for (int r = 0; r < 8; ++r) {      // D element r -> row r + 8*half, col mn
        const int row = row0 + r + 8 * half;
        h[(long)row * DD + col0 + mn] = acc[r] + bv;
    }
}

// per-node: beta = softmax(x @ Wc^T + bc) ; al/ar = <hh, attn> per (m, head)
// one wave per node, 8 waves per block
__global__ __launch_bounds__(256) void node_prep(const float* __restrict__ x,
                                                 const float* __restrict__ h,
                                                 const float* __restrict__ Wc,
                                                 const float* __restrict__ bc,
                                                 const float* __restrict__ attn_l,
                                                 const float* __restrict__ attn_r,
                                                 float* __restrict__ beta,
                                                 float* __restrict__ al,
                                                 float* __restrict__ ar) {
    const int lane = threadIdx.x & 31;
    const int node = blockIdx.x * 8 + (threadIdx.x >> 5);
    if (node >= NN) return;

    float p0 = 0.f, p1 = 0.f, p2 = 0.f;
    for (int j = lane; j < FF; j += 32) {
        const float xv = x[(long)node * FF + j];
        p0 += xv * Wc[0 * FF + j];
        p1 += xv * Wc[1 * FF + j];
        p2 += xv * Wc[2 * FF + j];
    }
    #pragma unroll
    for (int off = 16; off; off >>= 1) {
        p0 += __shfl_xor(p0, off, 32);
        p1 += __shfl_xor(p1, off, 32);
        p2 += __shfl_xor(p2, off, 32);
    }
    if (lane == 0) {
        p0 += bc[0]; p1 += bc[1]; p2 += bc[2];
        const float mx = fmaxf(p0, fmaxf(p1, p2));
        const float e0 = __expf(p0 - mx), e1 = __expf(p1 - mx), e2 = __expf(p2 - mx);
        const float inv = 1.0f / (e0 + e1 + e2);
        beta[node * 3 + 0] = e0 * inv;
        beta[node * 3 + 1] = e1 * inv;
        beta[node * 3 + 2] = e2 * inv;
    }

    #pragma unroll
    for (int hd = 0; hd < HH; ++hd) {
        const float hv = h[(long)node * DD + hd * CC + lane];
        #pragma unroll
        for (int m = 0; m < 2; ++m) {
            float dl = hv * attn_l[(m * HH + hd) * CC + lane];
            float dr = hv * attn_r[(m * HH + hd) * CC + lane];
            #pragma unroll
            for (int off = 16; off; off >>= 1) {
                dl += __shfl_xor(dl, off, 32);
                dr += __shfl_xor(dr, off, 32);
            }
            if (lane == 0) {
                al[(long)m * NN * HH + (long)node * HH + hd] = dl;
                ar[(long)m * NN * HH + (long)node * HH + hd] = dr;
            }
        }
    }
}

// pass 1: a_e = (ar[src] + al[dst]) * act ; segment max via u32 atomicMax
__global__ void edge_max(const int* __restrict__ ei,
                         const float* __restrict__ al,
                         const float* __restrict__ ar,
                         const float* __restrict__ alpha_act, int m,
                         float* __restrict__ abuf,
                         unsigned* __restrict__ amax) {
    const long e = (long)blockIdx.x * blockDim.x + threadIdx.x;
    if (e >= EE) return;
    const float act = alpha_act[m];
    const int dst = ei[e];
    const int src = ei[EE + e];
    #pragma unroll
    for (int hd = 0; hd < HH; ++hd) {
        const float a = (ar[(long)src * HH + hd] + al[(long)dst * HH + hd]) * act;
        abuf[e * HH + hd] = a;
        atomicMax(&amax[(long)dst * HH + hd], enc_f32(a));
    }
}

// pass 2: ex = exp(a - amax[dst]) ; segment sum
__global__ void edge_expsum(const int* __restrict__ ei,
                            const unsigned* __restrict__ amax,
                            float* __restrict__ abuf,
                            float* __restrict__ asum) {
    const long e = (long)blockIdx.x * blockDim.x + threadIdx.x;
    if (e >= EE) return;
    const int dst = ei[e];
    #pragma unroll
    for (int hd = 0; hd < HH; ++hd) {
        const float a  = abuf[e * HH + hd];
        const float mx = dec_f32(amax[(long)dst * HH + hd]);
        const float ex = __expf(a - mx);
        abuf[e * HH + hd] = ex;
        atomicAddF32(&asum[(long)dst * HH + hd], ex);
    }
}

// pass 3: out[dst] += beta[dst][m] * (ex/denom) * h[src]
// 2 edges per 256-thread block; 128 threads cover D=128 contiguous channels
__global__ __launch_bounds__(256) void edge_scatter(const int* __restrict__ ei,
                                                    const float* __restrict__ abuf,
                                                    const float* __restrict__ asum,
                                                    const float* __restrict__ beta, int m,
                                                    const float* __restrict__ h,
                                                    float* __restrict__ out) {
    const long e = (long)blockIdx.x * 2 + (threadIdx.x >> 7);
    if (e >= EE) return;
    const int t  = threadIdx.x & 127;  // channel within D
    const int hd = t >> 5;             // head
    const int dst = ei[e];
    const int src = ei[EE + e];
    const float ex    = abuf[e * HH + hd];
    const float denom = asum[(long)dst * HH + hd];
    const float w     = (ex / denom) * beta[(long)dst * 3 + m];
    const float v     = w * h[(long)src * DD + t];
    atomicAddF32(&out[(long)dst * DD + t], v);
}

// out = relu(out_acc + beta[:,2] * h)
__global__ void finalize(const float* __restrict__ h,
                         const float* __restrict__ beta,
                         float* __restrict__ out) {
    const long i = (long)blockIdx.x * blockDim.x + threadIdx.x;
    if (i >= (long)NN * DD) return;
    const long n = i / DD;
    const float v = out[i] + beta[n * 3 + 2] * h[i];
    out[i] = fmaxf(v, 0.0f);
}

// ---------------- launch ----------------

extern "C" void kernel_launch(void* const* d_in, const int* in_sizes, int n_in,
                              void* d_out, int out_size, void* d_ws, size_t ws_size,
                              hipStream_t stream) {
    const float* x         = (const float*)d_in[0];
    const int*   ei0       = (const int*)  d_in[1];
    const int*   ei1       = (const int*)  d_in[2];
    const float* W_lin     = (const float*)d_in[3];
    const float* b_lin     = (const float*)d_in[4];
    const float* attn_l    = (const float*)d_in[5];
    const float* attn_r    = (const float*)d_in[6];
    const float* alpha_act = (const float*)d_in[7];
    const float* W_conv    = (const float*)d_in[8];
    const float* b_conv    = (const float*)d_in[9];
    float* out = (float*)d_out;

    // workspace layout (floats)
    float* ws = (float*)d_ws;
    size_t off = 0;
    float*    h_buf = ws + off;             off += (size_t)NN * DD;      // 12.8M
    float*    beta  = ws + off;             off += (size_t)NN * 3;       // 0.3M
    float*    al    = ws + off;             off += (size_t)2 * NN * HH;  // 0.8M
    float*    ar    = ws + off;             off += (size_t)2 * NN * HH;  // 0.8M
    unsigned* amax  = (unsigned*)(ws + off);off += (size_t)2 * NN * HH;  // 0.8M
    float*    asum  = ws + off;             off += (size_t)2 * NN * HH;  // 0.8M (contiguous after amax)
    float*    abuf  = ws + off;             off += (size_t)EE * HH;      // 4.0M

    const long nd = (long)NN * DD;
    const long nstats = 4L * NN * HH;   // amax + asum together

    zero_u32<<<(int)((nd + 255) / 256), 256, 0, stream>>>((unsigned*)out, nd);
    zero_u32<<<(int)((nstats + 255) / 256), 256, 0, stream>>>(amax, nstats);

    gemm_h<<<NN / 16, 256, 0, stream>>>(x, W_lin, b_lin, h_buf);

    node_prep<<<NN / 8, 256, 0, stream>>>(x, h_buf, W_conv, b_conv,
                                          attn_l, attn_r, beta, al, ar);

    const int eblocks = (EE + 255) / 256;
    for (int m = 0; m < 2; ++m) {
        const int* ei = (m == 0) ? ei0 : ei1;
        const long so = (long)m * NN * HH;
        edge_max   <<<eblocks, 256, 0, stream>>>(ei, al + so, ar + so,
                                                 alpha_act, m, abuf, amax + so);
        edge_expsum<<<eblocks, 256, 0, stream>>>(ei, amax + so, abuf, asum + so);
        edge_scatter<<<EE / 2, 256, 0, stream>>>(ei, abuf, asum + so, beta, m,
                                                 h_buf, out);
    }

    finalize<<<(int)((nd + 255) / 256), 256, 0, stream>>>(h_buf, beta, out);
}